// PlainObj_9045201125961
// MI455X (gfx1250) — compile-verified
//
#include <hip/hip_runtime.h>
#include <hip/hip_bf16.h>
#include <math.h>

typedef float v2f __attribute__((ext_vector_type(2)));
typedef float v8f __attribute__((ext_vector_type(8)));

#define FACES   512
#define FSTRIDE 26          // floats per face record
#define NPIX    (224*224)
#define NBLOCKS 196         // 196*256 == 50176 == NPIX exactly
#define BIGF    3.0e38f

// ---------------- ws layout (float offsets) ----------------
// [0, 13312)            face records  (512*26)
// [13312, 14212)        tverts (300*3)
// [14212, 14516)        rowmin (304, as uint bits)
// [14516, 16564)        colmin (2048, as uint bits)
// [16564, 16760)        partial seg sums (196)
// [16760, 16956)        partial dpt sums (196)

// ============================================================
// Kernel 1: Rodrigues + transform + projection + per-face consts
// ============================================================
__global__ __launch_bounds__(512)
void setup_kernel(const float* __restrict__ rotvec, const float* __restrict__ trans,
                  const float* __restrict__ verts, const int* __restrict__ fidx,
                  float* __restrict__ wface, float* __restrict__ wtv,
                  unsigned* __restrict__ minbuf)
{
    const int t = threadIdx.x;

    // --- Rodrigues rotation (every thread recomputes; 9 scalar values) ---
    float rx = rotvec[0], ry = rotvec[1], rz = rotvec[2];
    float th = sqrtf(rx*rx + ry*ry + rz*rz);
    float it = 1.0f / th;
    float kx = rx*it, ky = ry*it, kz = rz*it;
    float sn = sinf(th), cs = cosf(th), oc = 1.0f - cs;
    float R00 = 1.0f + oc*(-(ky*ky + kz*kz));
    float R01 = -sn*kz + oc*(kx*ky);
    float R02 =  sn*ky + oc*(kx*kz);
    float R10 =  sn*kz + oc*(kx*ky);
    float R11 = 1.0f + oc*(-(kx*kx + kz*kz));
    float R12 = -sn*kx + oc*(ky*kz);
    float R20 = -sn*ky + oc*(kx*kz);
    float R21 =  sn*kx + oc*(ky*kz);
    float R22 = 1.0f + oc*(-(kx*kx + ky*ky));
    float tx = trans[0], ty = trans[1], tz = trans[2];

    // --- transformed vertices for EMD ---
    if (t < 300) {
        float vx = verts[t*3+0], vy = verts[t*3+1], vz = verts[t*3+2];
        wtv[t*3+0] = R00*vx + R01*vy + R02*vz + tx;
        wtv[t*3+1] = R10*vx + R11*vy + R12*vz + ty;
        wtv[t*3+2] = R20*vx + R21*vy + R22*vz + tz;
    }

    // --- per-face projected verts + rasterizer constants ---
    {
        float X[3], Y[3], Z[3];
        #pragma unroll
        for (int k = 0; k < 3; ++k) {
            int vi = fidx[t*3 + k];
            float vx = verts[vi*3+0], vy = verts[vi*3+1], vz = verts[vi*3+2];
            float wx = R00*vx + R01*vy + R02*vz + tx;
            float wy = R10*vx + R11*vy + R12*vz + ty;
            float wz = R20*vx + R21*vy + R22*vz + tz;
            X[k] =  wx / wz;
            Y[k] = -wy / wz;
            Z[k] =  wz;
        }
        float* fd = &wface[t*FSTRIDE];
        fd[0] = X[0]; fd[1] = Y[0]; fd[2] = X[1]; fd[3] = Y[1]; fd[4] = X[2]; fd[5] = Y[2];
        fd[6] = Z[0]; fd[7] = Z[1]; fd[8] = Z[2];
        fd[9]  = 1.0f/Z[0]; fd[10] = 1.0f/Z[1]; fd[11] = 1.0f/Z[2];
        float e0 = Y[1]-Y[2], e1 = X[2]-X[1], e2 = Y[2]-Y[0], e3 = X[0]-X[2];
        float den = e0*(X[0]-X[2]) + e1*(Y[0]-Y[2]);
        if (fabsf(den) < 1e-12f) den = 1e-12f;
        fd[12] = e0; fd[13] = e1; fd[14] = e2; fd[15] = e3; fd[16] = 1.0f/den;
        // edges (0->1), (1->2), (2->0): dir + 1/L2
        float ex, ey;
        ex = X[1]-X[0]; ey = Y[1]-Y[0];
        fd[17] = ex; fd[18] = ey; fd[19] = 1.0f/(ex*ex + ey*ey + 1e-12f);
        ex = X[2]-X[1]; ey = Y[2]-Y[1];
        fd[20] = ex; fd[21] = ey; fd[22] = 1.0f/(ex*ex + ey*ey + 1e-12f);
        ex = X[0]-X[2]; ey = Y[0]-Y[2];
        fd[23] = ex; fd[24] = ey; fd[25] = 1.0f/(ex*ex + ey*ey + 1e-12f);
    }

    // --- init row/col min buffers (304 + 2048) to FLT_MAX bits ---
    for (int i = t; i < 304 + 2048; i += 512) minbuf[i] = 0x7f7fffffu;
}

// ============================================================
// Kernel 2: soft rasterization over 224x224 crop, 512 faces
// ============================================================
__device__ __forceinline__ float edge_d2(float px, float py, float ax, float ay,
                                         float ex, float ey, float iL)
{
    float tp = ((px-ax)*ex + (py-ay)*ey) * iL;
    tp = fminf(fmaxf(tp, 0.0f), 1.0f);
    float dx = px - (ax + tp*ex);
    float dy = py - (ay + tp*ey);
    return dx*dx + dy*dy;
}

__global__ __launch_bounds__(256)
void raster_kernel(const float* __restrict__ wface,
                   const float* __restrict__ seg, const float* __restrict__ dpt,
                   const float* __restrict__ msk,
                   float* __restrict__ out_seg, float* __restrict__ out_dep,
                   float* __restrict__ pseg, float* __restrict__ pdpt)
{
    __shared__ float sf[FACES*FSTRIDE];
    __shared__ float red[256];
    const int tid = threadIdx.x;

    for (int i = tid; i < FACES*FSTRIDE; i += 256) sf[i] = wface[i];
    __syncthreads();

    const int p = blockIdx.x*256 + tid;             // exact: 196*256 == NPIX
    const int h = p / 224, w = p - h*224;
    const float px = (2.0f*(float)w + 1.0f - 256.0f) * (1.0f/256.0f);
    const float py = (255.0f - 2.0f*(float)h) * (1.0f/256.0f);

    float aprod = 1.0f;        // prod(1 - D)
    float zm    = 0.0f;        // running zmax (floor 0, matches reference)
    float S     = 0.0f;        // sum wgt
    float Nacc  = 0.0f;        // sum wgt * interpolated z

    for (int f = 0; f < FACES; ++f) {
        const float* fd = &sf[f*FSTRIDE];
        float x0=fd[0], y0=fd[1], x1=fd[2], y1=fd[3], x2=fd[4], y2=fd[5];
        float dx2 = px - x2, dy2 = py - y2;
        float w0 = (fd[12]*dx2 + fd[13]*dy2) * fd[16];
        float w1 = (fd[14]*dx2 + fd[15]*dy2) * fd[16];
        float w2 = 1.0f - w0 - w1;
        bool inside = (w0 >= 0.0f) & (w1 >= 0.0f) & (w2 >= 0.0f);

        float d2 = edge_d2(px, py, x0, y0, fd[17], fd[18], fd[19]);
        d2 = fminf(d2, edge_d2(px, py, x1, y1, fd[20], fd[21], fd[22]));
        d2 = fminf(d2, edge_d2(px, py, x2, y2, fd[23], fd[24], fd[25]));

        float sx = (inside ? 1.0f : -1.0f) * d2 * 1.0e5f;   // d2/SIGMA
        float D  = 1.0f / (1.0f + expf(-sx));

        float wc0 = fminf(fmaxf(w0, 0.0f), 1.0f);
        float wc1 = fminf(fmaxf(w1, 0.0f), 1.0f);
        float wc2 = fminf(fmaxf(w2, 0.0f), 1.0f);
        float wsum = wc0 + wc1 + wc2 + 1e-12f;
        float invw = 1.0f / wsum;
        float zden = wc0*fd[9] + wc1*fd[10] + wc2*fd[11];
        float zp   = wsum / zden;
        bool valid = (zp > 0.5f) & (zp < 100.0f);
        D = valid ? D : 0.0f;
        aprod *= (1.0f - D);

        float zn  = fminf(fmaxf((100.0f - zp) * (1.0f/99.5f), 0.0f), 1.0f);
        float zne = valid ? zn : 0.0f;
        if (zne > zm) {                       // online softmax rescale
            float sc = expf((zm - zne) * 1.0e4f);
            S *= sc; Nacc *= sc; zm = zne;
        }
        float e   = valid ? expf((zn - zm) * 1.0e4f) : 0.0f;
        float wgt = D * e;
        S += wgt;
        float cz = (wc0*fd[6] + wc1*fd[7] + wc2*fd[8]) * invw;
        Nacc += wgt * cz;
    }

    float wbg   = expf(-zm * 1.0e4f);
    float color = Nacc / (S + wbg);
    float alpha = 1.0f - aprod;

    out_seg[p] = alpha;
    out_dep[p] = color;

    float m0v = msk[p*3+0], m1v = msk[p*3+1], m2v = msk[p*3+2];
    float segl = fabsf(alpha - seg[p]) * m0v;
    float diff = color - dpt[p]*alpha;          // identical across channels
    float dptl = diff*diff * (m0v + m1v + m2v);

    red[tid] = segl; __syncthreads();
    for (int s2 = 128; s2 > 0; s2 >>= 1) { if (tid < s2) red[tid] += red[tid+s2]; __syncthreads(); }
    if (tid == 0) pseg[blockIdx.x] = red[0];
    __syncthreads();
    red[tid] = dptl; __syncthreads();
    for (int s2 = 128; s2 > 0; s2 >>= 1) { if (tid < s2) red[tid] += red[tid+s2]; __syncthreads(); }
    if (tid == 0) pdpt[blockIdx.x] = red[0];
}

// ============================================================
// Kernel 3: EMD pairwise distances via dual WMMA f32 16x16x4
//   E [i,j] = |a_i|^2 - 2 a_i.b_j   (norm folded into K=4 slot)
//   E'[j,i] = |b_j|^2 - 2 b_j.a_i   (transposed tile)
//   d2[i,j] = E[i,j] + |b_j|^2  =  E'[j,i] + |a_i|^2
//   column-reduce each tile per lane -> col mins of E give min over i,
//   col mins of E' give min over j. No cross-lane row butterflies needed.
// ============================================================
__global__ __launch_bounds__(32)
void emd_kernel(const float* __restrict__ tv, const float* __restrict__ pcd,
                unsigned* __restrict__ rowmin, unsigned* __restrict__ colmin)
{
    const int lane = threadIdx.x;
    const int m0 = blockIdx.y * 16;
    const int n0 = blockIdx.x * 16;
    const int lo = lane & 15;
    const bool hi = lane >= 16;

    int mm = m0 + lo;
    float ax = 0.0f, ay = 0.0f, az = 0.0f;
    if (mm < 300) { ax = tv[mm*3+0]; ay = tv[mm*3+1]; az = tv[mm*3+2]; }
    int nn = n0 + lo;                       // always < 2048
    float bx = pcd[nn*3+0], by = pcd[nn*3+1], bz = pcd[nn*3+2];

    float ra = ax*ax + ay*ay + az*az;
    float rb = bx*bx + by*by + bz*bz;

    // WMMA #1: rows = a-points (M), cols = b-points (N)
    //   A row_i = (-2ax, -2ay, -2az, ra_i) ; B col_j = (bx, by, bz, 1)
    v2f A1, B1;
    A1.x = hi ? (-2.0f*az) : (-2.0f*ax);  A1.y = hi ? ra   : (-2.0f*ay);
    B1.x = hi ? bz          : bx;         B1.y = hi ? 1.0f : by;

    // WMMA #2 (transposed): rows = b-points, cols = a-points
    //   A row_j = (-2bx, -2by, -2bz, rb_j) ; B col_i = (ax, ay, az, 1)
    v2f A2, B2;
    A2.x = hi ? (-2.0f*bz) : (-2.0f*bx);  A2.y = hi ? rb   : (-2.0f*by);
    B2.x = hi ? az          : ax;         B2.y = hi ? 1.0f : ay;

    v8f z1 = {0.f,0.f,0.f,0.f,0.f,0.f,0.f,0.f};
    v8f z2 = {0.f,0.f,0.f,0.f,0.f,0.f,0.f,0.f};
    v8f acc1 = __builtin_amdgcn_wmma_f32_16x16x4_f32(
        false, A1, false, B1, (short)0, z1, false, false);
    v8f acc2 = __builtin_amdgcn_wmma_f32_16x16x4_f32(
        false, A2, false, B2, (short)0, z2, false, false);

    // ---- col min of E (min over rows i) -> min over a-points for this lane's n
    float cm = BIGF;
    #pragma unroll
    for (int r = 0; r < 8; ++r) {
        int mg = m0 + r + (hi ? 8 : 0);     // row (a index) for this accumulator
        float q = (mg < 300) ? acc1[r] : BIGF;
        cm = fminf(cm, q);
    }
    cm = fminf(cm, __shfl_xor(cm, 16, 32)); // combine half-wave row groups
    float dcol = sqrtf(fmaxf(cm + rb, 0.0f) + 1e-12f);
    if (!hi) atomicMin(&colmin[nn], __float_as_uint(dcol));  // positive floats: uint order

    // ---- col min of E' (min over rows j) -> min over b-points for this lane's m
    float rm = BIGF;
    #pragma unroll
    for (int r = 0; r < 8; ++r)             // all b rows valid (2048 exact)
        rm = fminf(rm, acc2[r]);
    rm = fminf(rm, __shfl_xor(rm, 16, 32));
    float drow = sqrtf(fmaxf(rm + ra, 0.0f) + 1e-12f);
    if (!hi && mm < 300) atomicMin(&rowmin[mm], __float_as_uint(drow));
}

// ============================================================
// Kernel 4: final deterministic reductions -> scalars
// ============================================================
__global__ __launch_bounds__(256)
void final_kernel(const unsigned* __restrict__ rowmin, const unsigned* __restrict__ colmin,
                  const float* __restrict__ pseg, const float* __restrict__ pdpt,
                  float* __restrict__ out)
{
    __shared__ float red[256];
    const int t = threadIdx.x;

    auto block_sum = [&](float v) -> float {
        red[t] = v; __syncthreads();
        for (int s2 = 128; s2 > 0; s2 >>= 1) { if (t < s2) red[t] += red[t+s2]; __syncthreads(); }
        float r = red[0]; __syncthreads();
        return r;
    };

    float a = 0.0f; for (int i = t; i < 300;     i += 256) a += __uint_as_float(rowmin[i]);
    float rowsum = block_sum(a);
    float b = 0.0f; for (int i = t; i < 2048;    i += 256) b += __uint_as_float(colmin[i]);
    float colsum = block_sum(b);
    float c = 0.0f; for (int i = t; i < NBLOCKS; i += 256) c += pseg[i];
    float segsum = block_sum(c);
    float d = 0.0f; for (int i = t; i < NBLOCKS; i += 256) d += pdpt[i];
    float dptsum = block_sum(d);

    if (t == 0) {
        out[0] = 0.5f * (rowsum * (1.0f/300.0f) + colsum * (1.0f/2048.0f));
        out[1] = segsum * (1.0f/(float)NPIX);
        out[2] = dptsum * (1.0f/((float)NPIX*3.0f));
    }
}

// ============================================================
extern "C" void kernel_launch(void* const* d_in, const int* in_sizes, int n_in,
                              void* d_out, int out_size, void* d_ws, size_t ws_size,
                              hipStream_t stream)
{
    const float* rotvec = (const float*)d_in[0];
    const float* trans  = (const float*)d_in[1];
    const float* verts  = (const float*)d_in[2];
    const int*   fidx   = (const int*)  d_in[3];
    const float* seg    = (const float*)d_in[4];
    const float* dpt    = (const float*)d_in[5];
    const float* msk    = (const float*)d_in[6];
    const float* pcd    = (const float*)d_in[7];
    float* out = (float*)d_out;

    float* ws       = (float*)d_ws;
    float* wface    = ws;                       // 512*26
    float* wtv      = ws + 13312;               // 300*3
    unsigned* minbuf= (unsigned*)(ws + 14212);  // 304 + 2048
    unsigned* rowmin= minbuf;
    unsigned* colmin= minbuf + 304;
    float* pseg     = ws + 16564;               // 196
    float* pdpt     = ws + 16760;               // 196

    setup_kernel<<<1, 512, 0, stream>>>(rotvec, trans, verts, fidx, wface, wtv, minbuf);
    raster_kernel<<<NBLOCKS, 256, 0, stream>>>(wface, seg, dpt, msk,
                                               out + 3, out + 3 + NPIX, pseg, pdpt);
    dim3 gemd(128, 19);
    emd_kernel<<<gemd, 32, 0, stream>>>(wtv, pcd, rowmin, colmin);
    final_kernel<<<1, 256, 0, stream>>>(rowmin, colmin, pseg, pdpt, out);
}